// SSM_38165079392829
// MI455X (gfx1250) — compile-verified
//
#include <hip/hip_runtime.h>
#include <math.h>

// Problem constants (from reference)
#define BATCH 32
#define CPD   64
#define TDIM  4096
#define INDIM 256
#define HDIM  128
#define WDIM  128
#define TT    32            // time-tile length
#define NTILES (TDIM / TT)  // 128
#define CTRL_PITCH 33       // padded LDS row pitch for ctrl tile (TDM pad)
#define HS_PITCH   130      // hs row pitch: even -> 8B-aligned v2f A-fragment loads
#define PAIR_STRIDE 288     // interleaved B-matrix group stride (mod 64 == 32)

typedef __attribute__((ext_vector_type(2))) float    v2f;
typedef __attribute__((ext_vector_type(8))) float    v8f;
typedef __attribute__((ext_vector_type(4))) unsigned v4u;
typedef __attribute__((ext_vector_type(8))) int      v8i;
typedef __attribute__((ext_vector_type(4))) int      v4i;

// Native tanh if available on gfx1250 (v_tanh_f32), else ocml
#if __has_builtin(__builtin_amdgcn_tanhf)
  #define FAST_TANH(x) __builtin_amdgcn_tanhf(x)
#elif __has_builtin(__builtin_amdgcn_tanh_f32)
  #define FAST_TANH(x) __builtin_amdgcn_tanh_f32(x)
#else
  #define FAST_TANH(x) tanhf(x)
#endif

#if __has_builtin(__builtin_amdgcn_tensor_load_to_lds) && \
    __has_builtin(__builtin_amdgcn_s_wait_tensorcnt)
  #define USE_TDM 1
#endif

// ---------------------------------------------------------------------------
// Kernel A: Wcomb[c][h] = sum_i proj[c][i] * w_ih[h][i]   (64 x 128, K=256)
// ---------------------------------------------------------------------------
__global__ void SSM_wcomb_kernel(const float* __restrict__ proj,   // (64,256)
                                 const float* __restrict__ w_ih,   // (128,256)
                                 float* __restrict__ wcomb) {      // (64,128)
    int idx = blockIdx.x * blockDim.x + threadIdx.x;   // 0..8191
    int c = idx >> 7;
    int h = idx & 127;
    const float* p = proj + (size_t)c * INDIM;
    const float* w = w_ih + (size_t)h * INDIM;
    float a0 = 0.f, a1 = 0.f, a2 = 0.f, a3 = 0.f;      // 4 chains
#pragma unroll 4
    for (int i = 0; i < INDIM; i += 4) {
        a0 = fmaf(p[i + 0], w[i + 0], a0);
        a1 = fmaf(p[i + 1], w[i + 1], a1);
        a2 = fmaf(p[i + 2], w[i + 2], a2);
        a3 = fmaf(p[i + 3], w[i + 3], a3);
    }
    wcomb[idx] = (a0 + a1) + (a2 + a3);
}

// ---------------------------------------------------------------------------
// Kernel B: fused  xw -> scan -> out  per batch. One block (8 waves) / batch.
// B matrices are stored pair-interleaved in LDS: element (k, n) at
//   [(k>>1)*PAIR_STRIDE + 2n + (k&1)]
// so each WMMA B fragment (rows kb, kb+1; kb even) is one aligned ds_load_b64.
// ---------------------------------------------------------------------------
__global__ __launch_bounds__(256) void SSM_main_kernel(
    const float* __restrict__ control,  // (B, CPD, T)
    const float* __restrict__ w_hh,     // (H, H)
    const float* __restrict__ w_out,    // (W, H)
    const float* __restrict__ wcomb,    // (CPD, H)
    float* __restrict__ out)            // (B, T*W)
{
    __shared__ alignas(16) float lds_wout[(HDIM / 2) * PAIR_STRIDE];  // whh^T stage, then w_out^T pairs
    __shared__ alignas(16) float lds_wcomb[(CPD / 2) * PAIR_STRIDE];  // Wcomb pairs
    __shared__ float lds_xw[TT * HDIM];                  // phase-1 result
    __shared__ alignas(16) float lds_hs[TT * HS_PITCH];  // scan outputs (pitch 130)
    __shared__ float lds_ctrl[2][CPD * CTRL_PITCH];      // double-buffered ctrl tiles
    __shared__ alignas(16) float lds_h[2][HDIM];         // ping-pong hidden state
    __shared__ float lds_part[HDIM];                     // split-K partial sums

    const int b     = blockIdx.x;
    const int tid   = threadIdx.x;
    const int wave  = tid >> 5;
    const int lane  = tid & 31;
    const int lhalf = lane >> 4;      // 0 or 1
    const int l16   = lane & 15;
    const int jj    = tid & 127;      // output index for scan
    const int khalf = tid >> 7;       // 0: k=[0,64)   1: k=[64,128)

    // ---- one-time: w_hh^T into LDS (coalesced global read), then split to VGPRs
    for (int i = tid; i < HDIM * HDIM; i += 256) {
        int j = i >> 7, k = i & 127;
        lds_wout[k * HDIM + j] = w_hh[i];            // whhT[k][j] (linear layout)
    }
    __syncthreads();

    float wreg[64];                                   // half of w_hh row jj
#pragma unroll
    for (int k = 0; k < 64; ++k)
        wreg[k] = lds_wout[(khalf * 64 + k) * HDIM + jj];   // conflict-free
    __syncthreads();

    // ---- one-time: w_out^T and Wcomb into LDS (pair-interleaved); h = 0
    for (int i = tid; i < WDIM * HDIM; i += 256) {
        int w = i >> 7, h = i & 127;                  // B[k=h][n=w] = w_out[w][h]
        lds_wout[(h >> 1) * PAIR_STRIDE + 2 * w + (h & 1)] = w_out[i];
    }
    for (int i = tid; i < CPD * HDIM; i += 256) {
        int c = i >> 7, h = i & 127;                  // B[k=c][n=h] = wcomb[c][h]
        lds_wcomb[(c >> 1) * PAIR_STRIDE + 2 * h + (c & 1)] = wcomb[i];
    }
    if (tid < HDIM) { lds_h[0][tid] = 0.0f; lds_h[1][tid] = 0.0f; }

    const float* ctrl_b = control + (size_t)b * CPD * TDIM;
    float* out_b = out + (size_t)b * TDIM * WDIM;

#ifdef USE_TDM
    // Issue async TDM load of a 32x64 fp32 tile (LDS pad -> 33-float pitch)
    auto issue_tdm = [&](int tile, int buf) {
        unsigned lds_addr = (unsigned)(uintptr_t)&lds_ctrl[buf][0];
        unsigned long long ga = (unsigned long long)(uintptr_t)(ctrl_b + tile * TT);
        v4u g0;
        g0.s0 = 1u;                                   // count=1, user descriptor
        g0.s1 = lds_addr;                             // lds_addr (bytes)
        g0.s2 = (unsigned)(ga & 0xFFFFFFFFu);         // global_addr[31:0]
        g0.s3 = (unsigned)((ga >> 32) & 0x01FFFFFFu)  // global_addr[56:32]
                | (2u << 30);                         // type = 2 ("image")
        v8i g1;
        g1.s0 = (int)((2u << 16)      // data_size = 4B
                    | (1u << 20)      // pad_enable
                    | (4u << 22));    // pad_interval: 32 DWORDs, pad_amount=1 DWORD
        g1.s1 = (int)((TDIM & 0xFFFF) << 16);         // tensor_dim0[15:0] = 4096
        g1.s2 = (int)(((unsigned)TDIM >> 16) | ((unsigned)CPD << 16)); // dim0 hi | dim1
        g1.s3 = (int)((unsigned)TT << 16);            // tile_dim0 = 32
        g1.s4 = (int)CPD;                             // tile_dim1 = 64
        g1.s5 = (int)TDIM;                            // tensor_dim0_stride = 4096
        g1.s6 = 0; g1.s7 = 0;
        v4i z4 = {0, 0, 0, 0};
#if defined(__clang_major__) && (__clang_major__ >= 23)
        v8i z8 = {0, 0, 0, 0, 0, 0, 0, 0};
        __builtin_amdgcn_tensor_load_to_lds(g0, g1, z4, z4, z8, 0);
#else
        __builtin_amdgcn_tensor_load_to_lds(g0, g1, z4, z4, 0);
#endif
    };
    if (wave == 0) issue_tdm(0, 0);                   // prime the pipeline
#endif
    __syncthreads();

    int pp = 0;   // current h buffer parity (uniform across block)

    for (int tile = 0; tile < NTILES; ++tile) {
        const int t0  = tile * TT;
        const int cur = tile & 1;

#ifdef USE_TDM
        // -------- Phase 0: async prefetch next tile, retire current tile
        if (wave == 0) {
            if (tile + 1 < NTILES) {
                issue_tdm(tile + 1, cur ^ 1);
                __builtin_amdgcn_s_wait_tensorcnt(1); // current tile complete
            } else {
                __builtin_amdgcn_s_wait_tensorcnt(0);
            }
        }
        __syncthreads();
#else
        // -------- Phase 0: synchronous staging + prefetch of next tile
        for (int i = tid; i < CPD * TT; i += 256) {
            int c = i >> 5, t = i & 31;
            lds_ctrl[cur][c * CTRL_PITCH + t] = ctrl_b[(size_t)c * TDIM + t0 + t];
        }
        if (tile + 1 < NTILES && tid < CPD) {
            __builtin_prefetch(&ctrl_b[(size_t)tid * TDIM + t0 + TT], 0, 0);
        }
        __syncthreads();
#endif
        const float* ctile = &lds_ctrl[cur][0];

        // -------- Phase 1: xw(TT x H) = ctrl_tile(TT x CPD) @ Wcomb(CPD x H)
        // M=32 (2 tiles), N=128 (8 tiles), K=64 (16 wmma steps); 16 MN tiles / 8 waves
#pragma unroll
        for (int sub = 0; sub < 2; ++sub) {
            int tileid = wave * 2 + sub;
            int mt = tileid >> 3, nt = tileid & 7;
            int m = mt * 16 + l16;                    // t-local row of A
            int n = nt * 16 + l16;                    // h column of B
            v8f acc = {0.f,0.f,0.f,0.f,0.f,0.f,0.f,0.f};
#pragma unroll
            for (int kt = 0; kt < CPD / 4; ++kt) {
                int kb = kt * 4 + 2 * lhalf;          // K per ISA A/B layout (even)
                v2f a;
                a.x = ctile[(kb + 0) * CTRL_PITCH + m];   // A[m][k] = ctrl[c=k][t=m]
                a.y = ctile[(kb + 1) * CTRL_PITCH + m];
                v2f bf = *(const v2f*)&lds_wcomb[(kb >> 1) * PAIR_STRIDE + 2 * n];
                acc = __builtin_amdgcn_wmma_f32_16x16x4_f32(
                        false, a, false, bf, (short)0, acc, false, false);
            }
#pragma unroll
            for (int r = 0; r < 8; ++r) {
                int mm = mt * 16 + r + 8 * lhalf;     // D row per C/D layout
                lds_xw[mm * HDIM + n] = acc[r];
            }
        }
        __syncthreads();

        // -------- Phase 2: sequential scan, TT steps, split-K over 256 threads
        // h_new[j] = tanh(xw[s][j] + sum_k w_hh[j][k] * h[k])
        for (int s = 0; s < TT; ++s) {
            // 4 independent FMA chains + float4 LDS loads -> short critical path
            const float4* hb4 = (const float4*)&lds_h[pp][khalf * 64];
            float q0 = 0.f, q1 = 0.f, q2 = 0.f, q3 = 0.f;
#pragma unroll
            for (int q = 0; q < 16; ++q) {
                float4 hv = hb4[q];
                q0 = fmaf(wreg[4 * q + 0], hv.x, q0);
                q1 = fmaf(wreg[4 * q + 1], hv.y, q1);
                q2 = fmaf(wreg[4 * q + 2], hv.z, q2);
                q3 = fmaf(wreg[4 * q + 3], hv.w, q3);
            }
            float part = (q0 + q1) + (q2 + q3);
            if (khalf) lds_part[jj] = part;           // upper half publishes
            __syncthreads();
            if (!khalf) {
                float acc = lds_xw[s * HDIM + jj] + part + lds_part[jj];
                float hn = FAST_TANH(acc);
                lds_h[pp ^ 1][jj] = hn;
                lds_hs[s * HS_PITCH + jj] = hn;
            }
            pp ^= 1;
            __syncthreads();
        }

        // -------- Phase 3: out(TT x W) = hs(TT x H) @ woutT(H x W), then sin
        // M=32 (2), N=128 (8), K=128 (32 wmma steps)
#pragma unroll
        for (int sub = 0; sub < 2; ++sub) {
            int tileid = wave * 2 + sub;
            int mt = tileid >> 3, nt = tileid & 7;
            int m = mt * 16 + l16;
            int n = nt * 16 + l16;
            v8f acc = {0.f,0.f,0.f,0.f,0.f,0.f,0.f,0.f};
#pragma unroll
            for (int kt = 0; kt < HDIM / 4; ++kt) {
                int kb = kt * 4 + 2 * lhalf;          // even
                // A pair 8B-aligned (HS_PITCH even, kb even): single ds_load_b64
                v2f a  = *(const v2f*)&lds_hs[m * HS_PITCH + kb];
                v2f bf = *(const v2f*)&lds_wout[(kb >> 1) * PAIR_STRIDE + 2 * n];
                acc = __builtin_amdgcn_wmma_f32_16x16x4_f32(
                        false, a, false, bf, (short)0, acc, false, false);
            }
#pragma unroll
            for (int r = 0; r < 8; ++r) {
                int mm = mt * 16 + r + 8 * lhalf;         // t-local
                out_b[(size_t)(t0 + mm) * WDIM + n] = __sinf(acc[r]);
            }
        }
        __syncthreads();
    }
}

// ---------------------------------------------------------------------------
extern "C" void kernel_launch(void* const* d_in, const int* in_sizes, int n_in,
                              void* d_out, int out_size, void* d_ws, size_t ws_size,
                              hipStream_t stream) {
    const float* control = (const float*)d_in[0];   // (32,64,4096)
    const float* proj    = (const float*)d_in[1];   // (64,256)
    const float* w_ih    = (const float*)d_in[2];   // (128,256)
    const float* w_hh    = (const float*)d_in[3];   // (128,128)
    const float* w_out   = (const float*)d_in[4];   // (128,128)
    float* out   = (float*)d_out;                   // (32,1,4096*128) fp32
    float* wcomb = (float*)d_ws;                    // 64*128 floats scratch

    SSM_wcomb_kernel<<<(CPD * HDIM) / 256, 256, 0, stream>>>(proj, w_ih, wcomb);
    SSM_main_kernel<<<BATCH, 256, 0, stream>>>(control, w_hh, w_out, wcomb, out);
}